// CWICLinear_13572096656153
// MI455X (gfx1250) — compile-verified
//
#include <hip/hip_runtime.h>
#include <math.h>

// CWIC striped-masked linear: 8 stripes of [2048 x 4096] x [4096 x 512] f32 GEMM
// using V_WMMA_F32_16X16X4_F32. B (weight) tiles stream via async global->LDS
// (ASYNCcnt split-barrier, double buffered); A tiles get the per-stripe mask
// applied in registers on the way into LDS. Active-param counting fused via
// exact f32 atomics (all contributions are integers < 2^24).

#define K_DIM    4096
#define N_OUT    4096
#define SSIZE    512
#define BROWS    2048          // 4*512 token rows
#define WG_M     128
#define WG_N     256
#define KC       32
#define ASTR     36            // LDS row stride (floats) for A tile
#define BSTR     272           // LDS row stride (floats) for B tile
#define NCHUNK   (K_DIM / KC)  // 128
#define Y_ELEMS  (BROWS * N_OUT)       // 8388608
#define THR_SCALE 64.0f                // sqrt(4096)
#define DENSE_VAL 16777216.0f          // 512*8*4096 = 2^24

typedef float v2f __attribute__((ext_vector_type(2)));
typedef float v8f __attribute__((ext_vector_type(8)));
typedef int   v4i __attribute__((vector_size(4 * sizeof(int))));

// ---- async global -> LDS (16B per lane), tracked by ASYNCcnt ----
__device__ __forceinline__ void async_copy_b128(const float* g, float* l)
{
#if __has_builtin(__builtin_amdgcn_global_load_async_to_lds_b128)
    __builtin_amdgcn_global_load_async_to_lds_b128(
        (__attribute__((address_space(1))) v4i*)(void*)g,
        (__attribute__((address_space(3))) v4i*)(void*)l, 0, 0);
#else
    unsigned loff = (unsigned)(size_t)(__attribute__((address_space(3))) void*)l;
    unsigned long long ga = (unsigned long long)(size_t)g;
    asm volatile("global_load_async_to_lds_b128 %0, %1, off"
                 :: "v"(loff), "v"(ga) : "memory");
#endif
}

__device__ __forceinline__ void wait_async_le8()
{
    asm volatile("s_wait_asynccnt 0x8" ::: "memory");
}
__device__ __forceinline__ void wait_async_0()
{
    asm volatile("s_wait_asynccnt 0x0" ::: "memory");
}

__global__ void __launch_bounds__(256)
cwic_init_kernel(float* __restrict__ out)
{
    int i = blockIdx.x * blockDim.x + threadIdx.x;
    if (i < BROWS) {
        out[Y_ELEMS + i]         = DENSE_VAL;  // dense_params
        out[Y_ELEMS + BROWS + i] = 0.0f;       // active_params (atomic target)
    }
}

__global__ void __launch_bounds__(256)
cwic_gemm_kernel(const float* __restrict__ x,
                 const float* __restrict__ w,
                 const float* __restrict__ bias,
                 const float* __restrict__ thr,
                 const float* __restrict__ mu,
                 const float* __restrict__ stdv,
                 float* __restrict__ y,
                 float* __restrict__ active)
{
    __shared__ float    ldsA[WG_M * ASTR];     // masked x tile [128][32+pad]
    __shared__ float    ldsB[2][KC * BSTR];    // weight tiles  [32][256+pad], dbl-buffered
    __shared__ float    ldsMu[K_DIM];          // mu, whole K
    __shared__ float    ldsThr[K_DIM];         // thresholds[stripe]*64*std, whole K
    __shared__ unsigned ldsCnt[WG_M];          // per-row active counts

    const int tid    = threadIdx.x;
    const int ntile  = blockIdx.x;            // 0..1  (N tile of 256 within stripe)
    const int mtile  = blockIdx.y;            // 0..15 (M tile of 128)
    const int stripe = blockIdx.z;            // 0..7
    const int b0     = mtile * WG_M;
    const int c0     = stripe * SSIZE + ntile * WG_N;   // output column base

    // ---- one-time preload of mu and effective thresholds for this stripe ----
    for (int i = tid; i < K_DIM; i += 256) {
        ldsMu[i]  = mu[i];
        ldsThr[i] = thr[stripe * K_DIM + i] * THR_SCALE * stdv[i];
    }
    if (tid < WG_M) ldsCnt[tid] = 0u;

    // ---- wave / lane decomposition (wave32) ----
    const int lane = tid & 31;
    const int wave = tid >> 5;
    const int wm   = (wave & 1) * 64;   // wave M offset within WG tile
    const int wn   = (wave >> 1) * 64;  // wave N offset within WG tile
    const int hl   = lane >> 4;         // lane half: selects K +0 / +2
    const int l15  = lane & 15;
    const int koff = hl * 2;

    // ---- staging bases (per-iteration deltas are compile-time consts) ----
    const int amBase = tid >> 3;        // A row base (it adds 32)
    const int akOff  = (tid & 7) * 4;   // A k offset within chunk
    const int btr    = tid >> 6;        // B k row base (it adds 4)
    const int bn4    = (tid & 63) * 4;  // B n offset

    const float* xRow = x + (size_t)(b0 + amBase) * K_DIM + akOff;   // + it*32*K_DIM + k0g
    float*       aDst = ldsA + amBase * ASTR + akOff;                // + it*32*ASTR
    const float* wRow = w + (size_t)btr * N_OUT + c0 + bn4;          // + (k0g + it*4)*N_OUT
    // LDS B dst per it: ldsB[buf] + (btr + it*4)*BSTR + bn4

    v8f acc[4][4];
    const v8f vzero = {0.f, 0.f, 0.f, 0.f, 0.f, 0.f, 0.f, 0.f};
#pragma unroll
    for (int mi = 0; mi < 4; mi++)
#pragma unroll
        for (int ni = 0; ni < 4; ni++) acc[mi][ni] = vzero;

    unsigned cnt[4] = {0u, 0u, 0u, 0u};
    float4 ax[4];

    // ---- prologue: async B chunk 0 -> buf0, A chunk 0 -> regs ----
#pragma unroll
    for (int it = 0; it < 8; it++)
        async_copy_b128(wRow + (size_t)it * 4 * N_OUT,
                        &ldsB[0][(btr + it * 4) * BSTR + bn4]);
#pragma unroll
    for (int it = 0; it < 4; it++)
        ax[it] = *reinterpret_cast<const float4*>(xRow + (size_t)it * 32 * K_DIM);

    __syncthreads();   // ldsMu/ldsThr/ldsCnt visible

    for (int kc = 0; kc < NCHUNK; kc++) {
        const int k0g = kc * KC;
        const float* Bbuf = ldsB[kc & 1];

        // ---- masked A write: xm = (|x-mu| > thr) ? x : mu ----
#pragma unroll
        for (int it = 0; it < 4; it++) {
            const int gk = k0g + akOff;
            float xv[4] = {ax[it].x, ax[it].y, ax[it].z, ax[it].w};
            float ov[4];
            unsigned c = 0;
#pragma unroll
            for (int j = 0; j < 4; j++) {
                float m_ = ldsMu[gk + j];
                float t_ = ldsThr[gk + j];
                bool  on = fabsf(xv[j] - m_) > t_;
                ov[j] = on ? xv[j] : m_;
                c += on ? 1u : 0u;
            }
            cnt[it] += c;
            *reinterpret_cast<float4*>(aDst + it * 32 * ASTR) =
                make_float4(ov[0], ov[1], ov[2], ov[3]);
        }

        // ---- issue next chunk: async B -> other buffer, A -> regs ----
        if (kc + 1 < NCHUNK) {
            const int kn = k0g + KC;
            float* Bnxt = ldsB[(kc + 1) & 1];
#pragma unroll
            for (int it = 0; it < 8; it++)
                async_copy_b128(wRow + (size_t)(kn + it * 4) * N_OUT,
                                &Bnxt[(btr + it * 4) * BSTR + bn4]);
#pragma unroll
            for (int it = 0; it < 4; it++)
                ax[it] = *reinterpret_cast<const float4*>(
                    xRow + (size_t)it * 32 * K_DIM + kn);
            wait_async_le8();   // retire current chunk's 8 asyncs; next 8 in flight
        } else {
            wait_async_0();
        }
        __syncthreads();        // A tile + this chunk's B tile ready WG-wide

        // ---- compute: 8 k-steps of 4, 16 WMMAs each ----
#pragma unroll
        for (int kk = 0; kk < KC; kk += 4) {
            v2f afr[4], bfr[4];
#pragma unroll
            for (int mi = 0; mi < 4; mi++) {
                // A 16x4 f32 frag: lanes0-15 K={kk,kk+1}; lanes16-31 K={kk+2,kk+3}
                int base = (wm + mi * 16 + l15) * ASTR + kk + koff;
                v2f a = {ldsA[base], ldsA[base + 1]};
                afr[mi] = a;
            }
#pragma unroll
            for (int ni = 0; ni < 4; ni++) {
                // B 4x16 f32 frag: rows K={kk,kk+1} / {kk+2,kk+3} by lane half
                int base = (kk + koff) * BSTR + wn + ni * 16 + l15;
                v2f b = {Bbuf[base], Bbuf[base + BSTR]};
                bfr[ni] = b;
            }
#pragma unroll
            for (int mi = 0; mi < 4; mi++)
#pragma unroll
                for (int ni = 0; ni < 4; ni++)
                    acc[mi][ni] = __builtin_amdgcn_wmma_f32_16x16x4_f32(
                        false, afr[mi], false, bfr[ni],
                        (short)0, acc[mi][ni], false, false);
        }

        __syncthreads();        // compute done before ldsA / B buffer reuse
    }

    // ---- active-param counts (only one N-tile per (stripe, M-tile) counts) ----
    if (ntile == 0) {
#pragma unroll
        for (int it = 0; it < 4; it++)
            atomicAdd(&ldsCnt[it * 32 + amBase], cnt[it]);
        __syncthreads();
        if (tid < WG_M)
            atomicAdd(&active[b0 + tid], 512.0f * (float)ldsCnt[tid]);
    }

    // ---- epilogue: bias add + store C (frag: vgpr r -> M=r+8*half, N=l15) ----
#pragma unroll
    for (int ni = 0; ni < 4; ni++) {
        const int col = c0 + wn + ni * 16 + l15;
        const float bv = bias[col];
#pragma unroll
        for (int mi = 0; mi < 4; mi++) {
            const int rbase = b0 + wm + mi * 16 + hl * 8;
#pragma unroll
            for (int r = 0; r < 8; r++)
                y[(size_t)(rbase + r) * N_OUT + col] = acc[mi][ni][r] + bv;
        }
    }
}

extern "C" void kernel_launch(void* const* d_in, const int* in_sizes, int n_in,
                              void* d_out, int out_size, void* d_ws, size_t ws_size,
                              hipStream_t stream)
{
    const float* x    = (const float*)d_in[0];
    const float* w    = (const float*)d_in[1];
    const float* bias = (const float*)d_in[2];
    const float* thr  = (const float*)d_in[3];
    const float* mu   = (const float*)d_in[4];
    const float* stdv = (const float*)d_in[5];
    float* out = (float*)d_out;

    // dense_params + zero active_params (ordered before GEMM atomics on stream)
    cwic_init_kernel<<<(BROWS + 255) / 256, 256, 0, stream>>>(out);

    dim3 g(2, 16, 8);   // (N tiles per stripe, M tiles, stripes)
    cwic_gemm_kernel<<<g, 256, 0, stream>>>(
        x, w, bias, thr, mu, stdv,
        out,                              // y at offset 0
        out + Y_ELEMS + BROWS);           // active_params region
}